// GraphModel_31095563223580
// MI455X (gfx1250) — compile-verified
//
#include <hip/hip_runtime.h>

// ---------------------------------------------------------------------------
// 2-layer GCN for MI455X (gfx1250, wave32).
//   h' = D^-1/2 (A+I) D^-1/2 (h W) + b  applied twice.
// GEMM uses V_WMMA_F32_16X16X4_F32 (exact fp32 matrix path — GEMM is only
// ~6.6 GFLOP total, so full precision costs nothing; the workload is bound by
// the L2-resident edge gather/scatter, not the matrix units).
// ---------------------------------------------------------------------------

typedef __attribute__((ext_vector_type(2))) float v2f;
typedef __attribute__((ext_vector_type(8))) float v8f;

#define N_NODES 100000
#define DFEAT   128
#define ROW_TILES (N_NODES / 16)   // 6250 exact (100000 = 16 * 6250)
#define LDS_STRIDE (DFEAT + 4)     // 132 floats: row stride % 64 banks != 0

// ---- degree / normalization -----------------------------------------------

__global__ void gcn_fill_ones(float* __restrict__ p, int n) {
  int i = blockIdx.x * blockDim.x + threadIdx.x;
  if (i < n) p[i] = 1.0f;                 // self-loop contributes 1 to deg
}

__global__ void gcn_deg_accum(const int* __restrict__ dst, int e,
                              float* __restrict__ deg) {
  int i = blockIdx.x * blockDim.x + threadIdx.x;
  if (i < e) atomicAdd(&deg[dst[i]], 1.0f);
}

__global__ void gcn_rsqrt_inplace(float* __restrict__ p, int n) {
  int i = blockIdx.x * blockDim.x + threadIdx.x;
  if (i < n) p[i] = rsqrtf(p[i]);         // deg >= 1 always (self loop)
}

// ---- dense GEMM: H[N,128] = X[N,128] * W[128,128] via WMMA f32 16x16x4 ----
// Block = 256 threads = 8 waves. Block handles one 16-row tile of X (staged
// in LDS with +4 padding for conflict-free column reads); wave w computes
// the 16-column tile starting at w*16. K loop: 32 WMMA 16x16x4 steps.
//
// Fragment layouts per CDNA5 ISA 7.12.2:
//   A 16x4 : lane<16 -> M=lane,    v0:K=k0+0, v1:K=k0+1
//            lane>=16-> M=lane-16, v0:K=k0+2, v1:K=k0+3
//   B 4x16 : v_i -> row K = k0 + i + 2*(lane>>4), col N = lane&15
//   C/D    : VGPR r -> M = r + 8*(lane>>4), N = lane&15
__global__ __launch_bounds__(256)
void gcn_gemm_xw(const float* __restrict__ X, const float* __restrict__ W,
                 float* __restrict__ H) {
  __shared__ float xs[16 * LDS_STRIDE];

  const int rt  = blockIdx.x;            // row tile
  const int tid = threadIdx.x;

  // Cooperative load of the 16x128 X tile (512 float4s, 2 per thread).
  {
    const float4* Xv = (const float4*)(X + (size_t)rt * 16 * DFEAT);
    int t4 = tid;
    #pragma unroll
    for (int rep = 0; rep < 2; ++rep, t4 += 256) {
      int row = t4 >> 5;                 // 32 float4 per row
      int c4  = t4 & 31;
      float4 v = Xv[(size_t)row * 32 + c4];
      ((float4*)(xs + row * LDS_STRIDE))[c4] = v;  // 528B row stride, 16B ok
    }
  }
  __syncthreads();

  const int wave = tid >> 5;             // 0..7 -> column tile
  const int lane = tid & 31;
  const int half = lane >> 4;            // hi/lo half-wave
  const int mn   = lane & 15;            // M index for A, N index for B/C/D
  const int colb = wave * 16;

  v8f c = {};
  #pragma unroll 4
  for (int k0 = 0; k0 < DFEAT; k0 += 4) {
    const int ka = k0 + 2 * half;
    v2f a, b;
    a.x = xs[mn * LDS_STRIDE + ka];
    a.y = xs[mn * LDS_STRIDE + ka + 1];
    b.x = W[(size_t)ka * DFEAT + colb + mn];
    b.y = W[(size_t)(ka + 1) * DFEAT + colb + mn];
    // 8 args: (neg_a, A, neg_b, B, c_mod, C, reuse_a, reuse_b)
    c = __builtin_amdgcn_wmma_f32_16x16x4_f32(false, a, false, b,
                                              (short)0, c, false, false);
  }

  float* Hp = H + ((size_t)rt * 16) * DFEAT + colb + mn;
  #pragma unroll
  for (int r = 0; r < 8; ++r) {
    Hp[(size_t)(r + half * 8) * DFEAT] = c[r];
  }
}

// ---- out[i,:] = bias + dinv[i]^2 * h[i,:]  (self-loop term + bias) --------

__global__ __launch_bounds__(256)
void gcn_init_out(const float* __restrict__ H, const float* __restrict__ dinv,
                  const float* __restrict__ bias, float* __restrict__ Out) {
  int i = blockIdx.x * blockDim.x + threadIdx.x;   // over N*D
  if (i < N_NODES * DFEAT) {
    int node = i >> 7;
    int d    = i & (DFEAT - 1);
    float di = dinv[node];
    Out[i] = bias[d] + di * di * H[i];
  }
}

// ---- edge scatter: Out[dst,:] += dinv[src]*dinv[dst] * H[src,:] -----------
// One wave per edge: lane reads float4 of H[src] (512B coalesced per wave,
// L2-resident) and issues 4 global_atomic_add_f32 into Out[dst].
__global__ __launch_bounds__(256)
void gcn_scatter(const int* __restrict__ src, const int* __restrict__ dst,
                 const float* __restrict__ dinv, const float* __restrict__ H,
                 float* __restrict__ Out, int e) {
  int edge = blockIdx.x * 8 + (threadIdx.x >> 5);
  if (edge >= e) return;
  int lane = threadIdx.x & 31;
  int s = src[edge];
  int t = dst[edge];
  float norm = dinv[s] * dinv[t];
  float4 v = ((const float4*)(H + (size_t)s * DFEAT))[lane];
  float* o = Out + (size_t)t * DFEAT + lane * 4;
  atomicAdd(o + 0, v.x * norm);
  atomicAdd(o + 1, v.y * norm);
  atomicAdd(o + 2, v.z * norm);
  atomicAdd(o + 3, v.w * norm);
}

// ---------------------------------------------------------------------------

extern "C" void kernel_launch(void* const* d_in, const int* in_sizes, int n_in,
                              void* d_out, int out_size, void* d_ws, size_t ws_size,
                              hipStream_t stream) {
  const float* x   = (const float*)d_in[0];
  const int*  edge = (const int*)d_in[1];
  const float* W1  = (const float*)d_in[2];
  const float* b1  = (const float*)d_in[3];
  const float* W2  = (const float*)d_in[4];
  const float* b2  = (const float*)d_in[5];
  float* out = (float*)d_out;

  const int E = in_sizes[1] / 2;          // edge_index is [2, E]
  const int* srcp = edge;
  const int* dstp = edge + E;

  // workspace: dinv[N] then h[N*D]  (~51.6 MB)
  float* dinv = (float*)d_ws;
  float* h    = dinv + N_NODES;

  const int ND = N_NODES * DFEAT;
  dim3 blk(256);

  // degrees (shared by both layers): deg = 1 + in-degree; dinv = rsqrt(deg)
  gcn_fill_ones<<<(N_NODES + 255) / 256, blk, 0, stream>>>(dinv, N_NODES);
  gcn_deg_accum<<<(E + 255) / 256, blk, 0, stream>>>(dstp, E, dinv);
  gcn_rsqrt_inplace<<<(N_NODES + 255) / 256, blk, 0, stream>>>(dinv, N_NODES);

  // ---- layer 1: out = scatter(dinv, x@W1) + b1 ----
  gcn_gemm_xw<<<ROW_TILES, blk, 0, stream>>>(x, W1, h);
  gcn_init_out<<<(ND + 255) / 256, blk, 0, stream>>>(h, dinv, b1, out);
  gcn_scatter<<<(E + 7) / 8, blk, 0, stream>>>(srcp, dstp, dinv, h, out, E);

  // ---- layer 2: out = scatter(dinv, out@W2) + b2 ----
  gcn_gemm_xw<<<ROW_TILES, blk, 0, stream>>>(out, W2, h);
  gcn_init_out<<<(ND + 255) / 256, blk, 0, stream>>>(h, dinv, b2, out);
  gcn_scatter<<<(E + 7) / 8, blk, 0, stream>>>(srcp, dstp, dinv, h, out, E);
}